// Decoder_56736517980334
// MI455X (gfx1250) — compile-verified
//
#include <hip/hip_runtime.h>
#include <hip/hip_bf16.h>

// Problem dims (fixed by the reference)
#define S_  10
#define T_  200
#define F_  50
#define V_  60000
#define KP  52          // F padded to multiple of 4 (WMMA f32 K-step = 4)
#define TP  208         // T padded to multiple of 16
#define NV  80          // voxels per block tile; 60000 = 750 * 80 exactly
#define NTILES_V 750
#define MT  13          // TP/16 tiles along T
#define NT  5           // NV/16 tiles along V
#define NJOBS (MT*NT)   // 65 WMMA output tiles per block

typedef float v2f __attribute__((ext_vector_type(2)));
typedef float v8f __attribute__((ext_vector_type(8)));

#define NEG_HALF_LOG2PI (-0.91893853320467274f)

// log N(x; mu, sigma)
__device__ __forceinline__ float normal_lp(float x, float mu, float sg) {
    float z = (x - mu) / sg;
    return fmaf(-0.5f * z, z, -__logf(sg) + NEG_HALF_LOG2PI);
}

// ---------------------------------------------------------------------------
// Main fused kernel: RBF factors -> f32 WMMA GEMM -> Gaussian log-lik partials
// grid = (750, 10), block = 256 (8 wave32)
// ---------------------------------------------------------------------------
__global__ __launch_bounds__(256) void tfa_main(
    const float* __restrict__ data,     // [T,V]
    const float* __restrict__ R,        // [V,3]
    const float* __restrict__ W,        // [S,T,F]
    const float* __restrict__ FC,       // [S,F,3]
    const float* __restrict__ FW,       // [S,F]
    const float* __restrict__ Snoise,   // [T,V]
    float* __restrict__ partials)       // [S, NTILES_V]
{
    const int s   = blockIdx.y;
    const int v0  = blockIdx.x * NV;
    const int tid = threadIdx.x;

    __shared__ float Fsh[KP][NV];    // B operand: Factors tile, K-major   (16.6 KB)
    __shared__ float Wsh[KP][TP];    // A operand: Weights[s], K-major     (43.3 KB)
    __shared__ float cxs[KP], cys[KP], czs[KP], c2s[KP], iews[KP];
    __shared__ float rxs[NV], rys[NV], rzs[NV], r2s[NV];
    __shared__ float red[256];

    // ---- stage per-factor params (zero-padded rows 50..51) ----
    if (tid < KP) {
        float cx = 0.f, cy = 0.f, cz = 0.f, iw = 0.f;
        if (tid < F_) {
            const float* c = FC + ((size_t)s * F_ + tid) * 3;
            cx = c[0]; cy = c[1]; cz = c[2];
            iw = __expf(-FW[s * F_ + tid]);      // 1/exp(width)
        }
        cxs[tid] = cx; cys[tid] = cy; czs[tid] = cz;
        c2s[tid] = cx*cx + cy*cy + cz*cz;
        iews[tid] = iw;
    }
    // ---- stage voxel positions for this tile ----
    if (tid < NV) {
        const float* r = R + (size_t)(v0 + tid) * 3;
        float rx = r[0], ry = r[1], rz = r[2];
        rxs[tid] = rx; rys[tid] = ry; rzs[tid] = rz;
        r2s[tid] = rx*rx + ry*ry + rz*rz;
    }
    // ---- stage Weights[s] K-major, zero padded (k>=50 or t>=200 -> 0) ----
    for (int idx = tid; idx < KP * TP; idx += 256) {
        int k = idx / TP, t = idx % TP;
        float w = 0.f;
        if (t < T_ && k < F_) w = W[((size_t)s * T_ + t) * F_ + k];
        Wsh[k][t] = w;
    }
    __syncthreads();

    // ---- RBF factors tile: Fsh[f][v] = exp(-d2 / exp(width)) ----
    for (int idx = tid; idx < KP * NV; idx += 256) {
        int f = idx / NV, v = idx % NV;
        float val = 0.f;
        if (f < F_) {
            float d2 = r2s[v] - 2.f * (cxs[f]*rxs[v] + cys[f]*rys[v] + czs[f]*rzs[v]) + c2s[f];
            val = __expf(-d2 * iews[f]);
        }
        Fsh[f][v] = val;
    }
    __syncthreads();

    // ---- WMMA GEMM tiles + log-likelihood accumulation ----
    // f32 16x16x4 fragment layout (ISA 7.12.2):
    //   A (16x4): lane L -> row M = L%16; VGPR0 holds K = 2*(L>>4), VGPR1 K+1
    //   B (4x16): lane L -> col N = L%16; VGPR0 holds K = 2*(L>>4), VGPR1 K+1
    //   C (16x16): lane L, VGPR j -> (M = j + 8*(L>>4), N = L%16)
    const int wave  = tid >> 5;
    const int lane  = tid & 31;
    const int mrow  = lane & 15;
    const int khalf = (lane >> 4) << 1;          // 0 or 2

    float lpacc = 0.f;

    for (int job = wave; job < NJOBS; job += 8) {
        const int mt = job % MT, nt = job / MT;
        const int t0 = mt * 16, n0 = nt * 16;

        v8f c = {0.f,0.f,0.f,0.f,0.f,0.f,0.f,0.f};
        #pragma unroll
        for (int kk = 0; kk < KP; kk += 4) {
            v2f a, b;
            a.x = Wsh[kk + khalf    ][t0 + mrow];
            a.y = Wsh[kk + khalf + 1][t0 + mrow];
            b.x = Fsh[kk + khalf    ][n0 + mrow];
            b.y = Fsh[kk + khalf + 1][n0 + mrow];
            c = __builtin_amdgcn_wmma_f32_16x16x4_f32(
                    false, a, false, b, (short)0, c, false, false);
        }

        // consume Ymean tile: lp += log N(data; Ymean, Snoise)
        const int tbase = t0 + ((lane >> 4) << 3);   // +8 for upper lane half
        const int v     = v0 + n0 + mrow;
        #pragma unroll
        for (int j = 0; j < 8; ++j) {
            int t = tbase + j;
            if (t < T_) {
                size_t off = (size_t)t * V_ + v;
                lpacc += normal_lp(data[off], c[j], Snoise[off]);
            }
        }
    }

    // ---- deterministic block reduction ----
    red[tid] = lpacc;
    __syncthreads();
    for (int off = 128; off > 0; off >>= 1) {
        if (tid < off) red[tid] += red[tid + off];
        __syncthreads();
    }
    if (tid == 0) partials[(size_t)s * NTILES_V + blockIdx.x] = red[0];
}

// ---------------------------------------------------------------------------
// Finalize: per-s prior terms + deterministic reduction of tile partials
// grid = (10), block = 256
// ---------------------------------------------------------------------------
__global__ __launch_bounds__(256) void tfa_final(
    const float* __restrict__ partials,
    const float* __restrict__ W,    const float* __restrict__ FC,
    const float* __restrict__ FW,   const float* __restrict__ MW,
    const float* __restrict__ SW,   const float* __restrict__ MFC,
    const float* __restrict__ SFC,  const float* __restrict__ MFW,
    const float* __restrict__ SFW,  float* __restrict__ out)
{
    const int s = blockIdx.x, tid = threadIdx.x;
    __shared__ float red[256];
    float acc = 0.f;

    for (int i = tid; i < NTILES_V; i += 256) acc += partials[(size_t)s * NTILES_V + i];

    for (int i = tid; i < T_ * F_; i += 256)           // Weights prior [T,F]
        acc += normal_lp(W[(size_t)s * T_ * F_ + i], MW[i], SW[i]);
    for (int i = tid; i < F_ * 3; i += 256)            // FactorCenters prior [F,3]
        acc += normal_lp(FC[(size_t)s * F_ * 3 + i], MFC[i], SFC[i]);
    for (int i = tid; i < F_; i += 256)                // FactorWidths prior [F]
        acc += normal_lp(FW[(size_t)s * F_ + i], MFW[i], SFW[i]);

    red[tid] = acc;
    __syncthreads();
    for (int off = 128; off > 0; off >>= 1) {
        if (tid < off) red[tid] += red[tid + off];
        __syncthreads();
    }
    if (tid == 0) out[s] = red[0];
}

extern "C" void kernel_launch(void* const* d_in, const int* in_sizes, int n_in,
                              void* d_out, int out_size, void* d_ws, size_t ws_size,
                              hipStream_t stream) {
    const float* data   = (const float*)d_in[0];
    const float* R      = (const float*)d_in[1];
    const float* W      = (const float*)d_in[2];
    const float* FC     = (const float*)d_in[3];
    const float* FW     = (const float*)d_in[4];
    const float* MW     = (const float*)d_in[5];
    const float* SW     = (const float*)d_in[6];
    const float* MFC    = (const float*)d_in[7];
    const float* SFC    = (const float*)d_in[8];
    const float* MFW    = (const float*)d_in[9];
    const float* SFW    = (const float*)d_in[10];
    const float* Snoise = (const float*)d_in[11];

    float* partials = (float*)d_ws;   // S * NTILES_V = 7500 floats (30 KB)

    dim3 grid(NTILES_V, S_);
    tfa_main<<<grid, 256, 0, stream>>>(data, R, W, FC, FW, Snoise, partials);
    tfa_final<<<S_, 256, 0, stream>>>(partials, W, FC, FW, MW, SW,
                                      MFC, SFC, MFW, SFW, (float*)d_out);
}